// SpikingConv2D_71476845740373
// MI455X (gfx1250) — compile-verified
//
#include <hip/hip_runtime.h>

typedef __attribute__((ext_vector_type(2))) float v2f;
typedef __attribute__((ext_vector_type(8))) float v8f;

#define BDIM   256
#define WAVES  8      // 8 wave32 per block, one pixel per wave
#define KTOT   144    // 3*3*16 patch elements
#define NBLK   9      // 144 / 16 row-blocks
#define HH     56
#define WW     56
#define CC     16
#define FF     16

__global__ __launch_bounds__(BDIM) void spiking_conv2d_wmma_kernel(
    const float* __restrict__ tj,     // [4,56,56,16]
    const float* __restrict__ Jg,     // [3,3,16,16] == [144,16]
    const float* __restrict__ thr,    // [16]
    const float* __restrict__ alp,    // [16]
    float* __restrict__ out)          // [4,56,56,16]
{
    __shared__ float          Jt[KTOT * FF];          // weights, row k = di*48+dj*16+c
    __shared__ float          traw[WAVES][KTOT];      // raw patch times
    __shared__ float          tsort[WAVES][KTOT + 16];// sorted times (+sentinel)
    __shared__ unsigned short ordS[WAVES][KTOT + 16]; // argsort permutation

    const int tid  = threadIdx.x;
    const int w    = tid >> 5;
    const int lane = tid & 31;
    const int f    = lane & 15;   // feature handled by this lane
    const int hi   = lane >> 4;   // 0: rows 0-7 of C/D, 1: rows 8-15

    // cooperative load of weights into LDS
    for (int i = tid; i < KTOT * FF; i += BDIM) Jt[i] = Jg[i];

    const int p    = blockIdx.x * WAVES + w;   // pixel id 0..12543 (exact grid)
    const int bimg = p / (HH * WW);
    const int rem  = p % (HH * WW);
    const int y    = rem / WW;
    const int x    = rem % WW;

    // gather 3x3x16 patch ('same' padding with T_MIN = 0)
    for (int k = lane; k < KTOT; k += 32) {
        const int di = k / 48;
        const int dj = (k >> 4) % 3;
        const int c  = k & 15;
        const int yy = y + di - 1, xx = x + dj - 1;
        float v = 0.0f;
        if (yy >= 0 && yy < HH && xx >= 0 && xx < WW)
            v = tj[((bimg * HH + yy) * WW + xx) * CC + c];
        traw[w][k] = v;
    }
    __syncthreads();

    // stable rank sort: rank = #{j : t_j < t_k  or  (t_j == t_k and j < k)}
    for (int k = lane; k < KTOT; k += 32) {
        const float tk = traw[w][k];
        int rank = 0;
        for (int j = 0; j < KTOT; ++j) {
            const float tv = traw[w][j];
            rank += (tv < tk) || (tv == tk && j < k);
        }
        tsort[w][rank] = tk;
        ordS[w][rank]  = (unsigned short)k;
    }
    if (lane == 0) tsort[w][KTOT] = 1.0e6f;   // FAR_FUTURE sentinel for t_next
    __syncthreads();

    const float thrF = thr[f];
    const float alpF = alp[f];   // + alpha*T_MIN_PREV with T_MIN_PREV == 0

    // A operand: 16x16 lower-triangular ones, split into four 16x4 f32 tiles.
    // A layout (16x4 f32): lanes 0-15 M=0..15 hold {K0,K1}; lanes 16-31 hold {K2,K3}.
    v2f A[4];
#pragma unroll
    for (int i = 0; i < 4; ++i) {
        const int jx = 4 * i + 2 * hi;
        A[i].x = (jx     <= f) ? 1.0f : 0.0f;
        A[i].y = (jx + 1 <= f) ? 1.0f : 0.0f;
    }

    float carry_den = 0.0f, carry_num = 0.0f;  // prefix sums of completed blocks
    int   best_k  = 0x7fffffff;
    float best_ti = 0.0f;
    float ti0     = 0.0f;
    bool  found   = false;

    for (int blk = 0; blk < NBLK; ++blk) {
        v8f cd = {0.f,0.f,0.f,0.f,0.f,0.f,0.f,0.f};  // prefix of J rows
        v8f cn = {0.f,0.f,0.f,0.f,0.f,0.f,0.f,0.f};  // prefix of J*t rows
#pragma unroll
        for (int i = 0; i < 4; ++i) {
            // B operand (4x16 f32): lanes 0-15 hold rows {K0,K1}, lanes 16-31 {K2,K3}
            const int g0 = 16 * blk + 4 * i + 2 * hi;
            const int o0 = ordS[w][g0];
            const int o1 = ordS[w][g0 + 1];
            const float j0 = Jt[o0 * FF + f];
            const float j1 = Jt[o1 * FF + f];
            const float t0 = tsort[w][g0];
            const float t1 = tsort[w][g0 + 1];
            v2f bd; bd.x = j0;      bd.y = j1;
            v2f bn; bn.x = j0 * t0; bn.y = j1 * t1;
            // D = T16_tile x B + C  (f32 WMMA, full precision)
            cd = __builtin_amdgcn_wmma_f32_16x16x4_f32(false, A[i], false, bd,
                                                       (short)0, cd, false, false);
            cn = __builtin_amdgcn_wmma_f32_16x16x4_f32(false, A[i], false, bn,
                                                       (short)0, cn, false, false);
        }
        // scan this block's 16 prefix rows: lane(hi=0) owns k%16 in 0..7,
        // lane(hi=1) owns k%16 in 8..15 (C/D layout).
#pragma unroll
        for (int r = 0; r < 8; ++r) {
            const int   kg  = 16 * blk + 8 * hi + r;
            const float den = carry_den + cd[r] + alpF;   // J_sum + alpha
            const float num = carry_num + cn[r] + thrF;   // J_mult_tj + threshold
            const float ti  = num / den;
            const float tn  = tsort[w][kg + 1];
            const bool  valid = (ti < tn) && (den > 0.0f);
            if (kg == 0) ti0 = ti;                         // argmax-of-all-false fallback
            if (valid && !found) { found = true; best_k = kg; best_ti = ti; }
        }
        // carry = full block sum = row 15 (held in hi=1 lane's cd[7]/cn[7])
        const float d15 = cd[7], n15 = cn[7];
        const float d15x = __shfl_xor(d15, 16, 32);
        const float n15x = __shfl_xor(n15, 16, 32);
        carry_den += hi ? d15 : d15x;
        carry_num += hi ? n15 : n15x;
    }

    // combine the two half-wave candidates: earliest sorted index wins
    const int   ok  = __shfl_xor(best_k, 16, 32);
    const float oti = __shfl_xor(best_ti, 16, 32);
    if (ok < best_k) { best_k = ok; best_ti = oti; }
    float res = (best_k == 0x7fffffff) ? ti0 : best_ti;
    res = fminf(res, 1.0f);   // clamp to T_MAX
    if (hi == 0) out[p * FF + f] = res;
}

extern "C" void kernel_launch(void* const* d_in, const int* in_sizes, int n_in,
                              void* d_out, int out_size, void* d_ws, size_t ws_size,
                              hipStream_t stream) {
    const float* tj  = (const float*)d_in[0];   // [4,56,56,16]
    const float* ker = (const float*)d_in[1];   // [3,3,16,16]
    const float* thr = (const float*)d_in[2];   // [16]
    const float* alp = (const float*)d_in[3];   // [16]
    float* out = (float*)d_out;                 // [4,56,56,16]

    const int npix = 4 * HH * WW;               // 12544
    const int grid = npix / WAVES;              // 1568, exact
    spiking_conv2d_wmma_kernel<<<grid, BDIM, 0, stream>>>(tj, ker, thr, alp, out);
    (void)in_sizes; (void)n_in; (void)out_size; (void)d_ws; (void)ws_size;
}